// TSGNNDecoder_32822140076800
// MI455X (gfx1250) — compile-verified
//
#include <hip/hip_runtime.h>
#include <math.h>

typedef __attribute__((ext_vector_type(2))) float v2f;
typedef __attribute__((ext_vector_type(8))) float v8f;

#define NNODES 50000
#define NEDGES 1600000
#define WSTRIDE 100   // LDS row stride (avoids 64-bank conflicts: gcd(100,64) small)

// ---------------------------------------------------------------- utilities
__global__ void fill_kernel(float* __restrict__ p, int n, float v) {
    int i = blockIdx.x * blockDim.x + threadIdx.x;
    if (i < n) p[i] = v;
}

__global__ void deg_kernel(const long long* __restrict__ ei,
                           float* __restrict__ degc, float* __restrict__ degr, int E) {
    int e = blockIdx.x * blockDim.x + threadIdx.x;
    if (e >= E) return;
    long long r = ei[e];
    long long c = ei[(long long)E + e];
    atomicAdd(&degc[c], 1.0f);
    atomicAdd(&degr[r], 1.0f);
}

__global__ void inv_kernel(float* __restrict__ d, int n) {
    int i = blockIdx.x * blockDim.x + threadIdx.x;
    if (i >= n) return;
    float v = d[i];
    d[i] = (v > 0.0f) ? (1.0f / v) : 0.0f;
}

__global__ void mean_kernel(const float* __restrict__ a, const float* __restrict__ b,
                            const float* __restrict__ c, const float* __restrict__ d,
                            const float* __restrict__ e2, const float* __restrict__ f,
                            float* __restrict__ mu, float* __restrict__ md, int E) {
    int e = blockIdx.x * blockDim.x + threadIdx.x;
    if (e >= E) return;
    const float inv3 = 1.0f / 3.0f;
    mu[e] = (a[e] + b[e] + c[e]) * inv3;
    md[e] = (d[e] + e2[e] + f[e]) * inv3;
}

// ---------------------------------------------------------------- fused triple GEMM
// Computes [up|down|bias] = h_in @ [w_up;w_down;w_bias]^T  (96 output cols) using
// V_WMMA_F32_16X16X4_F32. One wave = 16 nodes x 96 cols (6 accumulators).
// Weights (96 x din) staged in LDS. up->upx, down->downx, bias->cat[:,64:96].
__global__ __launch_bounds__(256) void gemm_kernel(
    const float* __restrict__ hin, int din,
    const float* __restrict__ wup, const float* __restrict__ wdown,
    const float* __restrict__ wbias,
    float* __restrict__ upx, float* __restrict__ downx, float* __restrict__ cat,
    int nnodes)
{
    __shared__ float wl[96 * WSTRIDE];

    int tid = threadIdx.x;
    int tot = 96 * din;
    for (int i = tid; i < tot; i += 256) {
        int n = i / din;
        int k = i - n * din;
        float v;
        if (n < 32)      v = wup[n * din + k];
        else if (n < 64) v = wdown[(n - 32) * din + k];
        else             v = wbias[(n - 64) * din + k];
        wl[n * WSTRIDE + k] = v;
    }
    __syncthreads();

    int wave = tid >> 5;
    int lane = tid & 31;
    int ml   = lane & 15;     // M (for A) / N (for B,C,D) index within tile
    int hi   = lane >> 4;     // lane-half selects K pair {0,1} vs {2,3}
    int koff = hi << 1;

    int nodebase = blockIdx.x * 128 + wave * 16;
    int arow = nodebase + ml;
    if (arow >= nnodes) arow = nnodes - 1;        // clamp loads, keep EXEC all-ones
    const float* aptr = hin + (long long)arow * din;

    v8f z = {};
    v8f acc[6];
#pragma unroll
    for (int nt = 0; nt < 6; ++nt) acc[nt] = z;

    for (int k0 = 0; k0 < din; k0 += 4) {
        float2 av = *(const float2*)(aptr + k0 + koff);   // A[ml][k0+koff .. +1]
        v2f a; a.x = av.x; a.y = av.y;
#pragma unroll
        for (int nt = 0; nt < 6; ++nt) {
            // B[k][n] = W[n][k]; lane half picks K pair, ml picks column
            float2 bv = *(const float2*)(&wl[(nt * 16 + ml) * WSTRIDE + k0 + koff]);
            v2f b; b.x = bv.x; b.y = bv.y;
            acc[nt] = __builtin_amdgcn_wmma_f32_16x16x4_f32(
                false, a, false, b, (short)0, acc[nt], false, false);
        }
    }

#pragma unroll
    for (int nt = 0; nt < 6; ++nt) {
        int colb = nt * 16 + ml;
#pragma unroll
        for (int r = 0; r < 8; ++r) {
            int m = hi ? (8 + r) : r;                 // C/D layout: M=r / M=8+r
            int node = nodebase + m;
            if (node < nnodes) {
                float val = acc[nt][r];
                if (colb < 32)
                    upx[(long long)node * 32 + colb] = val;
                else if (colb < 64)
                    downx[(long long)node * 32 + (colb - 32)] = val;
                else
                    cat[(long long)node * 96 + colb] = val;
            }
        }
    }
}

// ---------------------------------------------------------------- edge scatter
// 4 threads per edge, 8 features each; both directions. Node arrays live in L2.
__global__ __launch_bounds__(256) void scatter_kernel(
    const long long* __restrict__ ei,
    const float* __restrict__ ewu, const float* __restrict__ ewd,
    const float* __restrict__ dinvc, const float* __restrict__ dinvr,
    const float* __restrict__ upx, const float* __restrict__ downx,
    float* __restrict__ cat, int E)
{
    int t = blockIdx.x * blockDim.x + threadIdx.x;
    int e = t >> 2;
    if (e >= E) return;
    int fo = (t & 3) << 3;

    long long r = ei[e];
    long long c = ei[(long long)E + e];
    float wu = dinvc[c] * ewu[e];
    float wd = dinvr[r] * ewd[e];

    const float4* us = (const float4*)(upx + r * 32 + fo);
    const float4* ds = (const float4*)(downx + c * 32 + fo);
    float4 u0 = us[0], u1 = us[1];
    float4 d0 = ds[0], d1 = ds[1];

    float* uo = cat + c * 96 + fo;        // x_up -> cols [0,32)
    float* od = cat + r * 96 + 32 + fo;   // x_down -> cols [32,64)

    atomicAdd(uo + 0, wu * u0.x); atomicAdd(uo + 1, wu * u0.y);
    atomicAdd(uo + 2, wu * u0.z); atomicAdd(uo + 3, wu * u0.w);
    atomicAdd(uo + 4, wu * u1.x); atomicAdd(uo + 5, wu * u1.y);
    atomicAdd(uo + 6, wu * u1.z); atomicAdd(uo + 7, wu * u1.w);

    atomicAdd(od + 0, wd * d0.x); atomicAdd(od + 1, wd * d0.y);
    atomicAdd(od + 2, wd * d0.z); atomicAdd(od + 3, wd * d0.w);
    atomicAdd(od + 4, wd * d1.x); atomicAdd(od + 5, wd * d1.y);
    atomicAdd(od + 6, wd * d1.z); atomicAdd(od + 7, wd * d1.w);
}

// ---------------------------------------------------------------- L2-normalize + leaky ReLU (in place)
__global__ __launch_bounds__(256) void norm_act_kernel(float* __restrict__ cat, int n) {
    int i = blockIdx.x * blockDim.x + threadIdx.x;
    if (i >= n) return;
    float* p = cat + (long long)i * 96;
    float ss = 0.0f;
#pragma unroll
    for (int j = 0; j < 24; ++j) {
        float4 v = ((const float4*)p)[j];
        ss += v.x * v.x + v.y * v.y + v.z * v.z + v.w * v.w;
    }
    float nrm = fmaxf(sqrtf(ss), 1e-12f);
    float inv = 1.0f / nrm;
#pragma unroll
    for (int j = 0; j < 24; ++j) {
        float4 v = ((float4*)p)[j];
        v.x *= inv; v.y *= inv; v.z *= inv; v.w *= inv;
        v.x = (v.x > 0.0f) ? v.x : 0.1f * v.x;
        v.y = (v.y > 0.0f) ? v.y : 0.1f * v.y;
        v.z = (v.z > 0.0f) ? v.z : 0.1f * v.z;
        v.w = (v.w > 0.0f) ? v.w : 0.1f * v.w;
        ((float4*)p)[j] = v;
    }
}

// ---------------------------------------------------------------- launch
extern "C" void kernel_launch(void* const* d_in, const int* in_sizes, int n_in,
                              void* d_out, int out_size, void* d_ws, size_t ws_size,
                              hipStream_t stream) {
    (void)in_sizes; (void)n_in; (void)out_size; (void)ws_size;

    const int N = NNODES;
    const int E = NEDGES;

    const float*     x  = (const float*)d_in[0];
    const long long* ei = (const long long*)d_in[1];
    const float* w1u = (const float*)d_in[2],  *w1d = (const float*)d_in[3],  *w1b = (const float*)d_in[4];
    const float* e1u = (const float*)d_in[5],  *e1d = (const float*)d_in[6];
    const float* w2u = (const float*)d_in[7],  *w2d = (const float*)d_in[8],  *w2b = (const float*)d_in[9];
    const float* e2u = (const float*)d_in[10], *e2d = (const float*)d_in[11];
    const float* w3u = (const float*)d_in[12], *w3d = (const float*)d_in[13], *w3b = (const float*)d_in[14];
    const float* e3u = (const float*)d_in[15], *e3d = (const float*)d_in[16];

    float* out = (float*)d_out;
    float* h_out = out;                       // [N,96]
    float* mu    = out + (size_t)N * 96;      // [E]
    float* md    = mu + E;                    // [E]

    // workspace layout (floats)
    float* ws     = (float*)d_ws;
    float* dinvc  = ws;                       // N   (in-deg at col, "up" direction)
    float* dinvr  = dinvc + N;                // N   (in-deg at row, "down" direction)
    float* upx    = dinvr + N;                // N*32
    float* downx  = upx + (size_t)N * 32;     // N*32
    float* h1     = downx + (size_t)N * 32;   // N*96
    float* h2     = h1 + (size_t)N * 96;      // N*96

    const int B = 256;
    dim3 blk(B);
    int gN96  = (N * 96 + B - 1) / B;
    int gE    = (E + B - 1) / B;
    int gE4   = (E * 4 + B - 1) / B;
    int gN    = (N + B - 1) / B;
    int gGemm = (N + 127) / 128;              // 128 nodes per block (8 waves x 16)

    // degrees (shared by all layers)
    fill_kernel<<<(2 * N + B - 1) / B, blk, 0, stream>>>(dinvc, 2 * N, 0.0f);
    deg_kernel<<<gE, blk, 0, stream>>>(ei, dinvc, dinvr, E);
    inv_kernel<<<(2 * N + B - 1) / B, blk, 0, stream>>>(dinvc, 2 * N);

    // edge-weight means
    mean_kernel<<<gE, blk, 0, stream>>>(e1u, e2u, e3u, e1d, e2d, e3d, mu, md, E);

    // ---- layer 1: x[N,8] -> h1[N,96]
    fill_kernel<<<gN96, blk, 0, stream>>>(h1, N * 96, 0.0f);
    gemm_kernel<<<gGemm, blk, 0, stream>>>(x, 8, w1u, w1d, w1b, upx, downx, h1, N);
    scatter_kernel<<<gE4, blk, 0, stream>>>(ei, e1u, e1d, dinvc, dinvr, upx, downx, h1, E);
    norm_act_kernel<<<gN, blk, 0, stream>>>(h1, N);

    // ---- layer 2: h1[N,96] -> h2[N,96]
    fill_kernel<<<gN96, blk, 0, stream>>>(h2, N * 96, 0.0f);
    gemm_kernel<<<gGemm, blk, 0, stream>>>(h1, 96, w2u, w2d, w2b, upx, downx, h2, N);
    scatter_kernel<<<gE4, blk, 0, stream>>>(ei, e2u, e2d, dinvc, dinvr, upx, downx, h2, E);
    norm_act_kernel<<<gN, blk, 0, stream>>>(h2, N);

    // ---- layer 3: h2[N,96] -> d_out[N,96] directly
    fill_kernel<<<gN96, blk, 0, stream>>>(h_out, N * 96, 0.0f);
    gemm_kernel<<<gGemm, blk, 0, stream>>>(h2, 96, w3u, w3d, w3b, upx, downx, h_out, N);
    scatter_kernel<<<gE4, blk, 0, stream>>>(ei, e3u, e3d, dinvc, dinvr, upx, downx, h_out, E);
    norm_act_kernel<<<gN, blk, 0, stream>>>(h_out, N);
}